// QuantNorm1D_new_52424370815659
// MI455X (gfx1250) — compile-verified
//
#include <hip/hip_runtime.h>
#include <cstdint>

// Problem constants (from reference setup_inputs)
#define F_DIM 1000
#define N_DIM 3000
#define B_DIM 64

// Tiling
#define TILE_F 64       // features per block
#define TILE_B 8        // batch rows per block (accumulators per thread)
#define CHUNK_N 40      // cdf_data rows staged per double-buffer stage
#define NCHUNK 75       // 3000 / 40
#define NSPLIT 4        // n-slices per block (threads 256 = 64 f * 4 n-slices)
#define ROWS_PT 10      // CHUNK_N / NSPLIT
#define LPS 5           // b64 async loads per thread per chunk: 40*64/2/256

// ---------------------------------------------------------------------------
// Branch-free erf (Abramowitz & Stegun 7.1.26, |err| <= 1.5e-7)
// ---------------------------------------------------------------------------
__device__ __forceinline__ float erf_fast(float z) {
  float ax = __builtin_fabsf(z);
  float t  = __builtin_amdgcn_rcpf(__builtin_fmaf(0.3275911f, ax, 1.0f));
  float p  = __builtin_fmaf(t, 1.061405429f, -1.453152027f);
  p = __builtin_fmaf(t, p, 1.421413741f);
  p = __builtin_fmaf(t, p, -0.284496736f);
  p = __builtin_fmaf(t, p, 0.254829592f);
  p *= t;
  float e = __expf(-ax * ax);                 // underflow -> 0 -> erf saturates to +/-1
  float r = __builtin_fmaf(-p, e, 1.0f);
  return __builtin_copysignf(r, z);
}

// ---------------------------------------------------------------------------
// Giles (2010) single-precision erfinv
// ---------------------------------------------------------------------------
__device__ __forceinline__ float erfinv_fast(float x) {
  float w = -__logf((1.0f - x) * (1.0f + x));
  float p;
  if (w < 5.0f) {
    w -= 2.5f;
    p = 2.81022636e-08f;
    p = __builtin_fmaf(p, w, 3.43273939e-07f);
    p = __builtin_fmaf(p, w, -3.5233877e-06f);
    p = __builtin_fmaf(p, w, -4.39150654e-06f);
    p = __builtin_fmaf(p, w, 0.00021858087f);
    p = __builtin_fmaf(p, w, -0.00125372503f);
    p = __builtin_fmaf(p, w, -0.00417768164f);
    p = __builtin_fmaf(p, w, 0.246640727f);
    p = __builtin_fmaf(p, w, 1.50140941f);
  } else {
    w = __builtin_sqrtf(w) - 3.0f;
    p = -0.000200214257f;
    p = __builtin_fmaf(p, w, 0.000100950558f);
    p = __builtin_fmaf(p, w, 0.00134934322f);
    p = __builtin_fmaf(p, w, -0.00367342844f);
    p = __builtin_fmaf(p, w, 0.00573950773f);
    p = __builtin_fmaf(p, w, -0.0076224613f);
    p = __builtin_fmaf(p, w, 0.00943887047f);
    p = __builtin_fmaf(p, w, 1.00167406f);
    p = __builtin_fmaf(p, w, 2.83297682f);
  }
  return p * x;
}

// ---------------------------------------------------------------------------
// gfx1250 async global->LDS staging of one CHUNK_N x TILE_F cdf_data tile.
// Each thread issues exactly LPS b64 loads (address-clamped, never predicated,
// so ASYNCcnt bookkeeping is exact per wave). Tracked by ASYNCcnt.
// ---------------------------------------------------------------------------
__device__ __forceinline__ void issue_chunk(const float* __restrict__ cdf,
                                            uint32_t lds_base, int c, int f0,
                                            int tid) {
  const uint64_t gbase = (uint64_t)(uintptr_t)cdf;
#pragma unroll
  for (int k = 0; k < LPS; ++k) {
    int p   = tid + k * 256;          // pair index within the 40x32-pair tile
    int row = p >> 5;                 // [0, 40)
    int col = (p & 31) * 2;           // even feature column [0, 64)
    // Clamp OOB feature columns (last f-tile) to a valid in-row address;
    // the garbage LDS slots are never read by valid compute lanes.
    int col_eff = (f0 + col + 1 < F_DIM) ? col : 0;
    uint64_t ga = gbase +
        4ull * ((uint64_t)(c * CHUNK_N + row) * (uint64_t)F_DIM +
                (uint64_t)(f0 + col_eff));
    uint32_t la = lds_base + (uint32_t)((row * TILE_F + col) * 4);
    asm volatile("global_load_async_to_lds_b64 %0, %1, off"
                 :: "v"(la), "v"(ga)
                 : "memory");
  }
}

extern "C" __global__ __launch_bounds__(256)
void kde_probit_kernel(const float* __restrict__ x,
                       const float* __restrict__ cdf,
                       const float* __restrict__ bwp,
                       float* __restrict__ out) {
  __shared__ float dbuf[2][CHUNK_N * TILE_F];           // 20 KB double buffer
  __shared__ float xs[TILE_B * TILE_F];                 // 2 KB query tile
  __shared__ float scl[TILE_F];                         // per-feature scale
  __shared__ float part[NSPLIT * TILE_B * TILE_F];      // 8 KB partial sums

  const int tid = threadIdx.x;
  const int f0  = blockIdx.x * TILE_F;
  const int b0  = blockIdx.y * TILE_B;
  const int tf  = tid & (TILE_F - 1);   // feature lane: 32 consecutive banks/wave
  const int tn  = tid >> 6;             // n-slice [0, 4)

  const uint32_t lds0 = (uint32_t)(uintptr_t)&dbuf[0][0];
  const uint32_t lds1 = (uint32_t)(uintptr_t)&dbuf[1][0];

  // Kick off chunk 0 immediately; overlap with x/scale staging below.
  issue_chunk(cdf, lds0, 0, f0, tid);

  // Stage query tile and per-feature scale s_f = 1/(sigmoid(p_f)*sqrt(2))
  //                                            = (1 + exp(-p_f)) / sqrt(2).
  for (int i = tid; i < TILE_B * TILE_F; i += 256) {
    int bl = i >> 6;
    int ff = f0 + (i & (TILE_F - 1));
    xs[i] = (ff < F_DIM) ? x[(b0 + bl) * F_DIM + ff] : 0.0f;
  }
  if (tid < TILE_F) {
    int ff = f0 + tid;
    float pb = (ff < F_DIM) ? bwp[ff] : 0.0f;
    scl[tid] = 0.70710678118654752f * (1.0f + __expf(-pb));
  }
  __syncthreads();

  const float sc = scl[tf];
  float xvs[TILE_B];
  float acc[TILE_B];
#pragma unroll
  for (int bl = 0; bl < TILE_B; ++bl) {
    xvs[bl] = xs[bl * TILE_F + tf] * sc;
    acc[bl] = 0.0f;
  }

  // Double-buffered pipeline over N. In-order ASYNC completion means that
  // after issuing the next stage's LPS loads, waiting for ASYNCcnt<=LPS
  // guarantees the current stage has fully landed in LDS.
  for (int c = 0; c < NCHUNK; ++c) {
    if (c + 1 < NCHUNK) {
      issue_chunk(cdf, (c & 1) ? lds0 : lds1, c + 1, f0, tid);
      asm volatile("s_wait_asynccnt 0x5" ::: "memory");   // == LPS
    } else {
      asm volatile("s_wait_asynccnt 0x0" ::: "memory");
    }
    __syncthreads();

    const float* db = (c & 1) ? &dbuf[1][0] : &dbuf[0][0];
#pragma unroll 2
    for (int r = 0; r < ROWS_PT; ++r) {
      float ds = db[(tn * ROWS_PT + r) * TILE_F + tf] * sc;  // conflict-free
#pragma unroll
      for (int bl = 0; bl < TILE_B; ++bl)
        acc[bl] += erf_fast(xvs[bl] - ds);
    }
    __syncthreads();   // current buffer free for the stage after next
  }

  // Cross-n-slice reduction + probit finalize.
#pragma unroll
  for (int bl = 0; bl < TILE_B; ++bl)
    part[(tn * TILE_B + bl) * TILE_F + tf] = acc[bl];
  __syncthreads();

#pragma unroll
  for (int i = 0; i < 2; ++i) {
    int idx = tid + i * 256;                    // [0, 512): bl*64 + tf
    float s = part[idx] + part[idx + 512] + part[idx + 1024] + part[idx + 1536];
    int bl = idx >> 6;
    int ff = f0 + (idx & (TILE_F - 1));
    if (ff < F_DIM) {
      // mean CDF = 0.5 + 0.5 * (sum_erf / N)
      float mean = __builtin_fmaf(s, 0.5f / (float)N_DIM, 0.5f);
      mean = fminf(fmaxf(mean, 9e-08f), 1.0f - 9e-08f);
      float u = __builtin_fmaf(2.0f, mean, -1.0f);
      out[(b0 + bl) * F_DIM + ff] = 1.41421356237309515f * erfinv_fast(u);
    }
  }
}

extern "C" void kernel_launch(void* const* d_in, const int* in_sizes, int n_in,
                              void* d_out, int out_size, void* d_ws, size_t ws_size,
                              hipStream_t stream) {
  (void)in_sizes; (void)n_in; (void)d_ws; (void)ws_size; (void)out_size;
  const float* x   = (const float*)d_in[0];   // (64, 1000)
  const float* cdf = (const float*)d_in[1];   // (3000, 1000)
  const float* bwp = (const float*)d_in[2];   // (1, 1000)
  float* out = (float*)d_out;                 // (64, 1000)

  dim3 grid((F_DIM + TILE_F - 1) / TILE_F, B_DIM / TILE_B, 1);  // 16 x 8
  kde_probit_kernel<<<grid, dim3(256, 1, 1), 0, stream>>>(x, cdf, bwp, out);
}